// DariushFlashAttention2_19533511262800
// MI455X (gfx1250) — compile-verified
//
#include <hip/hip_runtime.h>
#include <hip/hip_bf16.h>

// CDNA5 / gfx1250 wave32 WMMA implementation of the per-token head-mixing
// attention. One wave handles one token position; all Q/K/V fragments are
// fetched with disjoint per-lane b64/b32 global loads (zero redundancy),
// LDS is used only to transpose the 32x32 softmax weights from WMMA C-layout
// into A-layout for the second GEMM. Softmax row reductions use single
// ds_swizzle_b32 XOR butterflies (16-lane groups).

typedef __attribute__((ext_vector_type(2))) float v2f;
typedef __attribute__((ext_vector_type(8))) float v8f;

#define WAVES_PER_BLOCK 8
#define E_DIM 4096           // NUM_HEADS * KEY_SIZE
#define H_DIM 32
#define D_DIM 128
#define W_STRIDE 34          // padded LDS row stride (words); 34*4 bytes keeps b64 align

__device__ __forceinline__ v8f wmma_f32(v2f a, v2f b, v8f c) {
  // V_WMMA_F32_16X16X4_F32 : D = A(16x4) * B(4x16) + C(16x16)
  return __builtin_amdgcn_wmma_f32_16x16x4_f32(
      /*neg_a=*/false, a, /*neg_b=*/false, b,
      /*c_mod=*/(short)0, c, /*reuse_a=*/false, /*reuse_b=*/false);
}

// ds_swizzle_b32, group-of-32 mode: offset = xor_mask<<10 | or_mask<<5 | and_mask
// XOR masks 1/2/4/8 stay inside each 16-lane half, which is exactly the
// reduction domain of one 16-wide WMMA C-tile column group.
template <int XOR_MASK>
__device__ __forceinline__ float swz_xor(float x) {
  return __int_as_float(
      __builtin_amdgcn_ds_swizzle(__float_as_int(x), (XOR_MASK << 10) | 0x1F));
}

__device__ __forceinline__ float redmax16(float x) {
  x = fmaxf(x, swz_xor<8>(x));
  x = fmaxf(x, swz_xor<4>(x));
  x = fmaxf(x, swz_xor<2>(x));
  x = fmaxf(x, swz_xor<1>(x));
  return x;
}

__device__ __forceinline__ float redsum16(float x) {
  x += swz_xor<8>(x);
  x += swz_xor<4>(x);
  x += swz_xor<2>(x);
  x += swz_xor<1>(x);
  return x;
}

__global__ __launch_bounds__(WAVES_PER_BLOCK * 32)
void dariush_headmix_attn_kernel(const float* __restrict__ q,
                                 const float* __restrict__ k,
                                 const float* __restrict__ v,
                                 float* __restrict__ out,
                                 int n_tokens) {
  const int wave = threadIdx.x >> 5;
  const int lane = threadIdx.x & 31;
  const int half = lane >> 4;      // 0: lanes 0-15, 1: lanes 16-31
  const int l16  = lane & 15;

  const long tok = (long)blockIdx.x * WAVES_PER_BLOCK + wave;
  if (tok >= n_tokens) return;

  const float* __restrict__ qbase = q + tok * E_DIM;
  const float* __restrict__ kbase = k + tok * E_DIM;
  const float* __restrict__ vbase = v + tok * E_DIM;
  float* __restrict__ obase = out + tok * E_DIM;

  // V is consumed last (GEMM2): prefetch its 16KB now so it arrives under
  // the GEMM1 + softmax latency. Lane L covers row L; 4x128B lines per row.
  {
    const float* vrow = vbase + lane * D_DIM;
    __builtin_prefetch(vrow +  0, 0, 1);   // global_prefetch_b8
    __builtin_prefetch(vrow + 32, 0, 1);
    __builtin_prefetch(vrow + 64, 0, 1);
    __builtin_prefetch(vrow + 96, 0, 1);
  }

  // per-wave padded 32x32 weight tile for the C-layout -> A-layout transpose
  __shared__ float lds_w[WAVES_PER_BLOCK][H_DIM * W_STRIDE];
  float* __restrict__ wlds = lds_w[wave];

  // ---------------- GEMM 1: L[32,32] = Q[32,128] * K^T ----------------
  v8f acc[2][2];
  #pragma unroll
  for (int mi = 0; mi < 2; ++mi)
    #pragma unroll
    for (int ni = 0; ni < 2; ++ni)
      acc[mi][ni] = (v8f){0.f, 0.f, 0.f, 0.f, 0.f, 0.f, 0.f, 0.f};

  // A-frag (Q): lane(half,l16) -> Q[mi*16+l16][4*kk + 2*half .. +1]  (b64)
  // B-frag (K^T): lane         -> K[ni*16+l16][4*kk + 2*half .. +1]  (b64)
  const int coff_base = 2 * half;
  #pragma unroll 4
  for (int kk = 0; kk < 32; ++kk) {
    const int coff = kk * 4 + coff_base;
    v2f qa0 = *(const v2f*)(qbase + (l16)      * D_DIM + coff);
    v2f qa1 = *(const v2f*)(qbase + (16 + l16) * D_DIM + coff);
    v2f kb0 = *(const v2f*)(kbase + (l16)      * D_DIM + coff);
    v2f kb1 = *(const v2f*)(kbase + (16 + l16) * D_DIM + coff);
    acc[0][0] = wmma_f32(qa0, kb0, acc[0][0]);
    acc[0][1] = wmma_f32(qa0, kb1, acc[0][1]);
    acc[1][0] = wmma_f32(qa1, kb0, acc[1][0]);
    acc[1][1] = wmma_f32(qa1, kb1, acc[1][1]);
  }

  // ---------------- softmax over head axis g (32 values/row) ----------------
  // C layout: acc[mi][ni][j] @ lane -> row h = mi*16 + j + 8*half,
  //                                    col g = ni*16 + l16
  const float scale = 0.08838834764831845f; // 1/sqrt(128)
  #pragma unroll
  for (int mi = 0; mi < 2; ++mi) {
    #pragma unroll
    for (int j = 0; j < 8; ++j) {
      float a0 = acc[mi][0][j];
      float a1 = acc[mi][1][j];
      float m = redmax16(fmaxf(a0, a1));
      float p0 = __expf((a0 - m) * scale);
      float p1 = __expf((a1 - m) * scale);
      float inv = __builtin_amdgcn_rcpf(redsum16(p0 + p1));  // v_rcp_f32
      p0 *= inv;
      p1 *= inv;
      const int h = mi * 16 + j + 8 * half;
      wlds[h * W_STRIDE + l16]      = p0;
      wlds[h * W_STRIDE + 16 + l16] = p1;
    }
  }
  __syncthreads();  // LDS store->load visibility (8 identical waves, cheap)

  // Re-load W as A-layout fragments: lane -> W[mi*16+l16][4*kk + 2*half .. +1]
  v2f wa[2][8];
  #pragma unroll
  for (int mi = 0; mi < 2; ++mi)
    #pragma unroll
    for (int kk = 0; kk < 8; ++kk)
      wa[mi][kk] = *(const v2f*)(wlds + (mi * 16 + l16) * W_STRIDE + kk * 4 + coff_base);

  // ---------------- GEMM 2: O[32,128] = W[32,32] * V[32,128] ----------------
  #pragma unroll 2
  for (int ni = 0; ni < 8; ++ni) {
    const int col = ni * 16 + l16;
    // B-frag (V): lane -> V[4*kk + 2*half][col], V[4*kk + 2*half + 1][col]
    v2f vb[8];
    #pragma unroll
    for (int kk = 0; kk < 8; ++kk) {
      float x0 = vbase[(4 * kk + 2 * half)     * D_DIM + col];
      float x1 = vbase[(4 * kk + 2 * half + 1) * D_DIM + col];
      vb[kk] = (v2f){x0, x1};
    }
    v8f o0 = (v8f){0.f, 0.f, 0.f, 0.f, 0.f, 0.f, 0.f, 0.f};
    v8f o1 = (v8f){0.f, 0.f, 0.f, 0.f, 0.f, 0.f, 0.f, 0.f};
    #pragma unroll
    for (int kk = 0; kk < 8; ++kk) {
      o0 = wmma_f32(wa[0][kk], vb[kk], o0);
      o1 = wmma_f32(wa[1][kk], vb[kk], o1);
    }
    #pragma unroll
    for (int j = 0; j < 8; ++j) {
      obase[(j + 8 * half) * D_DIM + col]      = o0[j];
      obase[(16 + j + 8 * half) * D_DIM + col] = o1[j];
    }
  }
}

extern "C" void kernel_launch(void* const* d_in, const int* in_sizes, int n_in,
                              void* d_out, int out_size, void* d_ws, size_t ws_size,
                              hipStream_t stream) {
  const float* q = (const float*)d_in[0];
  const float* k = (const float*)d_in[1];
  const float* v = (const float*)d_in[2];
  float* out = (float*)d_out;

  const int n_tokens = in_sizes[0] / E_DIM;              // B*S = 8192
  const int blocks = (n_tokens + WAVES_PER_BLOCK - 1) / WAVES_PER_BLOCK;
  dariush_headmix_attn_kernel<<<blocks, WAVES_PER_BLOCK * 32, 0, stream>>>(
      q, k, v, out, n_tokens);
}